// DialogueGCNModel_76192719831070
// MI455X (gfx1250) — compile-verified
//
#include <hip/hip_runtime.h>
#include <hip/hip_bf16.h>

typedef __attribute__((ext_vector_type(2))) float f32x2;
typedef __attribute__((ext_vector_type(8))) float f32x8;

#define N_NODES 8192
#define F_IN 200
#define HDIM 100
#define NREL 8
#define N_BASES 30
#define N_CLASSES 6
#define D_EMO 300  // F_IN + HDIM
#define NKB (N_NODES / 16)  // 512 key blocks of 16

// V_WMMA_F32_16X16X4_F32: D(16x16 f32) = A(16x4 f32) * B(4x16 f32) + C
__device__ __forceinline__ f32x8 wmma_f32(f32x2 a, f32x2 b, f32x8 c) {
  return __builtin_amdgcn_wmma_f32_16x16x4_f32(false, a, false, b, (short)0, c,
                                               false, false);
}

// ---------------------------------------------------------------------------
// W[r,f,h] = sum_b comp[r,b] * basis[b,f,h]
__global__ void k_compute_W(const float* __restrict__ comp,
                            const float* __restrict__ basis,
                            float* __restrict__ W) {
  int idx = blockIdx.x * blockDim.x + threadIdx.x;
  if (idx >= NREL * F_IN * HDIM) return;
  int r = idx / (F_IN * HDIM);
  int fh = idx % (F_IN * HDIM);
  float acc = 0.f;
#pragma unroll
  for (int b = 0; b < N_BASES; ++b)
    acc += comp[r * N_BASES + b] * basis[(size_t)b * F_IN * HDIM + fh];
  W[idx] = acc;
}

__global__ void k_zero(float* __restrict__ p, int n) {
  int i = blockIdx.x * blockDim.x + threadIdx.x;
  if (i < n) p[i] = 0.f;
}

__global__ void k_deg(const int* __restrict__ dst, const int* __restrict__ et,
                      float* __restrict__ deg, int E) {
  int e = blockIdx.x * blockDim.x + threadIdx.x;
  if (e < E) atomicAdd(&deg[dst[e] * NREL + et[e]], 1.0f);
}

__global__ void k_inv(float* __restrict__ deg, int n) {
  int i = blockIdx.x * blockDim.x + threadIdx.x;
  if (i < n) {
    float d = deg[i];
    deg[i] = (d > 0.f) ? 1.0f / d : 0.f;
  }
}

// h1[dst] += xW[et, src] * inv[dst*R+et]   (one edge per block, lanes over H)
__global__ __launch_bounds__(128) void k_scatter_msg(
    const int* __restrict__ src, const int* __restrict__ dst,
    const int* __restrict__ et, const float* __restrict__ inv,
    const float* __restrict__ xW, float* __restrict__ h1) {
  int e = blockIdx.x;
  int h = threadIdx.x;
  if (h >= HDIM) return;
  int s = src[e], d = dst[e], t = et[e];
  float w = inv[d * NREL + t];
  atomicAdd(&h1[(size_t)d * HDIM + h],
            xW[((size_t)t * N_NODES + s) * HDIM + h] * w);
}

// agg[dst] += h1[src]
__global__ __launch_bounds__(128) void k_scatter_agg(
    const int* __restrict__ src, const int* __restrict__ dst,
    const float* __restrict__ h1, float* __restrict__ agg) {
  int e = blockIdx.x;
  int h = threadIdx.x;
  if (h >= HDIM) return;
  atomicAdd(&agg[(size_t)dst[e] * HDIM + h], h1[(size_t)src[e] * HDIM + h]);
}

__global__ void k_concat(const float* __restrict__ x,
                         const float* __restrict__ h2,
                         float* __restrict__ emo) {
  int i = blockIdx.x * blockDim.x + threadIdx.x;
  if (i >= N_NODES * D_EMO) return;
  int n = i / D_EMO, c = i % D_EMO;
  emo[i] = (c < F_IN) ? x[(size_t)n * F_IN + c]
                      : h2[(size_t)n * HDIM + (c - F_IN)];
}

// ---------------------------------------------------------------------------
// Generic WMMA fp32 GEMM: C[M x Nreal] = A[M x K] @ B[K x Nreal] (+bias)(+=C)(relu)
// grid = (ceil(Nreal/16), M/128, batch), block = 256 (8 waves, 16x16 tile/wave)
// Masked output columns are handled by CLAMPING the load column (garbage only
// reaches lanes whose output column is never stored) -> no EXEC churn in loop.
__global__ __launch_bounds__(256) void k_gemm(
    const float* __restrict__ A, const float* __restrict__ B,
    const float* __restrict__ bias, float* __restrict__ C, int K, int Nreal,
    int lda, int ldb, int ldc, long long sA, long long sB, long long sC,
    int flags) {
  const int tid = threadIdx.x;
  const int wave = tid >> 5;
  const int lane = tid & 31;
  const int half = lane >> 4;  // 0: rows m0..m0+7 / K 0,1 ; 1: rows +8 / K 2,3
  const int l16 = lane & 15;
  const int kk = half * 2;
  const int m0 = blockIdx.y * 128 + wave * 16;
  const int n0 = blockIdx.x * 16;
  A += (size_t)blockIdx.z * sA;
  B += (size_t)blockIdx.z * sB;
  C += (size_t)blockIdx.z * sC;
  const int n = n0 + l16;
  const bool nok = (n < Nreal);
  const int nc = nok ? n : (Nreal - 1);  // clamped load column

  f32x8 acc = {};
  const float* ap = A + (m0 + l16) * lda + kk;  // 32-bit index math (buffers < 2^31)
  const float* bp = B + kk * ldb + nc;
  const int bstep = 4 * ldb;
  for (int k0 = 0; k0 < K; k0 += 4) {
    float2 af = *(const float2*)ap;  // A[m, k0+kk .. +1], 8B aligned (lda even)
    ap += 4;
    f32x2 a;
    a[0] = af.x;
    a[1] = af.y;
    f32x2 b;
    b[0] = bp[0];    // B[k0+kk,   nc]
    b[1] = bp[ldb];  // B[k0+kk+1, nc]
    bp += bstep;
    acc = wmma_f32(a, b, acc);
  }
  if (nok) {
    float bi = (bias != nullptr) ? bias[n] : 0.f;
#pragma unroll
    for (int j = 0; j < 8; ++j) {
      int row = m0 + j + half * 8;
      size_t idx = (size_t)row * ldc + n;
      float v = acc[j] + bi;
      if (flags & 1) v += C[idx];
      if (flags & 2) v = fmaxf(v, 0.f);
      C[idx] = v;
    }
  }
}

// ---------------------------------------------------------------------------
// Flash-style attention: em2 = softmax(before @ emo^T) @ emo
// grid = N/16 blocks (16 query rows each), block = 256 (8 waves)
__global__ __launch_bounds__(256) void k_attention(
    const float* __restrict__ before, const float* __restrict__ emo,
    float* __restrict__ em2) {
  __shared__ float sQ[16 * D_EMO];    // query tile (rows of `before`)
  __shared__ float sP[8][16 * 16];    // exp(S - m) tiles, one per wave
  __shared__ float sM[16][8];
  __shared__ float sS[16][8];
  __shared__ float fM[16];
  __shared__ float fS[16];

  const int tid = threadIdx.x;
  const int wave = tid >> 5;
  const int lane = tid & 31;
  const int half = lane >> 4;
  const int l16 = lane & 15;
  const int kk = half * 2;
  const int q0 = blockIdx.x * 16;

  for (int i = tid; i < 16 * D_EMO; i += 256)
    sQ[i] = before[(size_t)q0 * D_EMO + i];
  __syncthreads();

  const float* qp = &sQ[l16 * D_EMO + kk];

  // ---- pass 1: per-row running max + sumexp (waves stride over key blocks)
  float m[8], s[8];
#pragma unroll
  for (int j = 0; j < 8; ++j) {
    m[j] = -3.0e38f;
    s[j] = 0.f;
  }
  for (int kb = wave; kb < NKB; kb += 8) {
    const int j0 = kb * 16;
    const float* ep = &emo[(size_t)(j0 + l16) * D_EMO + kk];
    f32x8 acc = {};
    for (int k0 = 0; k0 < D_EMO; k0 += 4) {
      float2 af = *(const float2*)(qp + k0);
      f32x2 a;
      a[0] = af.x;
      a[1] = af.y;
      float2 bf = *(const float2*)(ep + k0);
      f32x2 b;
      b[0] = bf.x;
      b[1] = bf.y;
      acc = wmma_f32(a, b, acc);
    }
#pragma unroll
    for (int j = 0; j < 8; ++j) {
      float v = acc[j];
      float bm = v;  // max over the 16 key columns (xor<16 stays in halves)
      bm = fmaxf(bm, __shfl_xor(bm, 1));
      bm = fmaxf(bm, __shfl_xor(bm, 2));
      bm = fmaxf(bm, __shfl_xor(bm, 4));
      bm = fmaxf(bm, __shfl_xor(bm, 8));
      float mn = fmaxf(m[j], bm);
      float p = __expf(v - mn);
      float ps = p;
      ps += __shfl_xor(ps, 1);
      ps += __shfl_xor(ps, 2);
      ps += __shfl_xor(ps, 4);
      ps += __shfl_xor(ps, 8);
      s[j] = s[j] * __expf(m[j] - mn) + ps;
      m[j] = mn;
    }
  }
  if (l16 == 0) {
#pragma unroll
    for (int j = 0; j < 8; ++j) {
      int row = j + half * 8;
      sM[row][wave] = m[j];
      sS[row][wave] = s[j];
    }
  }
  __syncthreads();
  if (tid < 16) {
    float gm = -3.0e38f;
    for (int w = 0; w < 8; ++w) gm = fmaxf(gm, sM[tid][w]);
    float gs = 0.f;
    for (int w = 0; w < 8; ++w) gs += sS[tid][w] * __expf(sM[tid][w] - gm);
    fM[tid] = gm;
    fS[tid] = gs;
  }
  __syncthreads();

  // ---- pass 2: O = sum_j exp(S - m) @ V ; wave owns V col tiles {wave, +8, +16}
  f32x8 oacc[3];
#pragma unroll
  for (int t = 0; t < 3; ++t) oacc[t] = (f32x8){};

  for (int ob = 0; ob < NKB / 8; ++ob) {
    // each wave: S for its own key block of this outer iteration -> P in LDS
    const int j0 = ob * 128 + wave * 16;
    const float* ep = &emo[(size_t)(j0 + l16) * D_EMO + kk];
    f32x8 acc = {};
    for (int k0 = 0; k0 < D_EMO; k0 += 4) {
      float2 af = *(const float2*)(qp + k0);
      f32x2 a;
      a[0] = af.x;
      a[1] = af.y;
      float2 bf = *(const float2*)(ep + k0);
      f32x2 b;
      b[0] = bf.x;
      b[1] = bf.y;
      acc = wmma_f32(a, b, acc);
    }
#pragma unroll
    for (int j = 0; j < 8; ++j) {
      int row = j + half * 8;  // query-local row
      sP[wave][row * 16 + l16] = __expf(acc[j] - fM[row]);
    }
    __syncthreads();
    // P @ V over the 8 P tiles of this outer block (clamped V columns)
    for (int pp = 0; pp < 8; ++pp) {
      const float* vp = &emo[(size_t)(ob * 128 + pp * 16 + kk) * D_EMO];
#pragma unroll
      for (int k0 = 0; k0 < 16; k0 += 4) {
        float2 af = *(const float2*)&sP[pp][l16 * 16 + k0 + kk];
        f32x2 a;
        a[0] = af.x;
        a[1] = af.y;
        const float* vk = vp + k0 * D_EMO;
#pragma unroll
        for (int t = 0; t < 3; ++t) {
          int vtile = wave + 8 * t;
          if (vtile >= 19) break;  // wave-uniform
          int col = vtile * 16 + l16;
          int colc = (col < D_EMO) ? col : (D_EMO - 1);  // clamp, no EXEC churn
          f32x2 b;
          b[0] = vk[colc];
          b[1] = vk[D_EMO + colc];
          oacc[t] = wmma_f32(a, b, oacc[t]);
        }
      }
    }
    __syncthreads();
  }
#pragma unroll
  for (int t = 0; t < 3; ++t) {
    int vtile = wave + 8 * t;
    if (vtile >= 19) break;
#pragma unroll
    for (int j = 0; j < 8; ++j) {
      int row = j + half * 8;
      int col = vtile * 16 + l16;
      if (col < D_EMO)
        em2[(size_t)(q0 + row) * D_EMO + col] = oacc[t][j] / fS[row];
    }
  }
}

// ---------------------------------------------------------------------------
__global__ void k_classify(const float* __restrict__ hid,
                           const float* __restrict__ w,
                           const float* __restrict__ b,
                           float* __restrict__ out) {
  int n = blockIdx.x * blockDim.x + threadIdx.x;
  if (n >= N_NODES) return;
  float lg[N_CLASSES];
#pragma unroll
  for (int c = 0; c < N_CLASSES; ++c) lg[c] = b[c];
  const float* h = hid + (size_t)n * HDIM;
  for (int k = 0; k < HDIM; ++k) {
    float hv = h[k];
#pragma unroll
    for (int c = 0; c < N_CLASSES; ++c) lg[c] += hv * w[k * N_CLASSES + c];
  }
  float mx = lg[0];
#pragma unroll
  for (int c = 1; c < N_CLASSES; ++c) mx = fmaxf(mx, lg[c]);
  float se = 0.f;
#pragma unroll
  for (int c = 0; c < N_CLASSES; ++c) se += __expf(lg[c] - mx);
  float lse = mx + __logf(se);
#pragma unroll
  for (int c = 0; c < N_CLASSES; ++c) out[(size_t)n * N_CLASSES + c] = lg[c] - lse;
}

// ---------------------------------------------------------------------------
extern "C" void kernel_launch(void* const* d_in, const int* in_sizes, int n_in,
                              void* d_out, int out_size, void* d_ws,
                              size_t ws_size, hipStream_t stream) {
  const float* x = (const float*)d_in[0];
  const int* eidx = (const int*)d_in[1];
  /* d_in[2] edge_norm: unused by reference */
  const int* etype = (const int*)d_in[3];
  const float* basis = (const float*)d_in[4];
  const float* comp = (const float*)d_in[5];
  const float* root_w = (const float*)d_in[6];
  const float* root_b = (const float*)d_in[7];
  const float* gc_rel_w = (const float*)d_in[8];
  const float* gc_rel_b = (const float*)d_in[9];
  const float* gc_root_w = (const float*)d_in[10];
  const float* beta_w = (const float*)d_in[11];
  const float* beta_b = (const float*)d_in[12];
  const float* lin_w = (const float*)d_in[13];
  const float* lin_b = (const float*)d_in[14];
  const float* smax_w = (const float*)d_in[15];
  const float* smax_b = (const float*)d_in[16];
  const int E = in_sizes[3];
  const int* src = eidx;
  const int* dst = eidx + E;

  float* ws = (float*)d_ws;
  size_t off = 0;
  auto alloc = [&](size_t nelem) {
    float* p = ws + off;
    off += (nelem + 63) & ~(size_t)63;
    return p;
  };
  float* W = alloc((size_t)NREL * F_IN * HDIM);      // 160000
  float* xW = alloc((size_t)NREL * N_NODES * HDIM);  // 6.55M
  float* inv = alloc((size_t)N_NODES * NREL);        // 65536
  float* h1 = alloc((size_t)N_NODES * HDIM);
  float* agg = alloc((size_t)N_NODES * HDIM);
  float* h2 = alloc((size_t)N_NODES * HDIM);
  float* emo = alloc((size_t)N_NODES * D_EMO);
  float* bef = alloc((size_t)N_NODES * D_EMO);
  float* em2 = alloc((size_t)N_NODES * D_EMO);
  float* hid = alloc((size_t)N_NODES * HDIM);
  (void)ws_size;
  (void)n_in;
  (void)out_size;

  const int MB = N_NODES / 128;  // grid.y for all GEMMs (M = 8192)

  // 1) basis combination
  k_compute_W<<<(NREL * F_IN * HDIM + 255) / 256, 256, 0, stream>>>(comp, basis, W);
  // 2) xW[r] = x @ W[r]   (batched over 8 relations)
  k_gemm<<<dim3(7, MB, NREL), 256, 0, stream>>>(
      x, W, nullptr, xW, F_IN, HDIM, F_IN, HDIM, HDIM, 0LL,
      (long long)F_IN * HDIM, (long long)N_NODES * HDIM, 0);
  // 3) per-(dst,rel) degree -> inverse
  k_zero<<<(N_NODES * NREL + 255) / 256, 256, 0, stream>>>(inv, N_NODES * NREL);
  k_deg<<<(E + 255) / 256, 256, 0, stream>>>(dst, etype, inv, E);
  k_inv<<<(N_NODES * NREL + 255) / 256, 256, 0, stream>>>(inv, N_NODES * NREL);
  // 4) h1 = x @ root_w + root_b, then scatter mean-aggregated messages on top
  k_gemm<<<dim3(7, MB, 1), 256, 0, stream>>>(x, root_w, root_b, h1, F_IN, HDIM,
                                             F_IN, HDIM, HDIM, 0LL, 0LL, 0LL, 0);
  k_scatter_msg<<<E, 128, 0, stream>>>(src, dst, etype, inv, xW, h1);
  // 5) agg = segment_sum(h1[src], dst)
  k_zero<<<(N_NODES * HDIM + 255) / 256, 256, 0, stream>>>(agg, N_NODES * HDIM);
  k_scatter_agg<<<E, 128, 0, stream>>>(src, dst, h1, agg);
  // 6) h2 = agg @ gc_rel_w + gc_rel_b + h1 @ gc_root_w
  k_gemm<<<dim3(7, MB, 1), 256, 0, stream>>>(agg, gc_rel_w, gc_rel_b, h2, HDIM,
                                             HDIM, HDIM, HDIM, HDIM, 0LL, 0LL,
                                             0LL, 0);
  k_gemm<<<dim3(7, MB, 1), 256, 0, stream>>>(h1, gc_root_w, nullptr, h2, HDIM,
                                             HDIM, HDIM, HDIM, HDIM, 0LL, 0LL,
                                             0LL, 1 /*accumulate*/);
  // 7) emotions = [x | h2]
  k_concat<<<(N_NODES * D_EMO + 255) / 256, 256, 0, stream>>>(x, h2, emo);
  // 8) before = emotions @ beta_w + beta_b
  k_gemm<<<dim3(19, MB, 1), 256, 0, stream>>>(emo, beta_w, beta_b, bef, D_EMO,
                                              D_EMO, D_EMO, D_EMO, D_EMO, 0LL,
                                              0LL, 0LL, 0);
  // 9) em2 = softmax(before @ emotions^T) @ emotions   (flash-style)
  k_attention<<<N_NODES / 16, 256, 0, stream>>>(bef, emo, em2);
  // 10) hidden = relu(em2 @ lin_w + lin_b)
  k_gemm<<<dim3(7, MB, 1), 256, 0, stream>>>(em2, lin_w, lin_b, hid, D_EMO,
                                             HDIM, D_EMO, HDIM, HDIM, 0LL, 0LL,
                                             0LL, 2 /*relu*/);
  // 11) logits + log_softmax
  k_classify<<<(N_NODES + 255) / 256, 256, 0, stream>>>(hid, smax_w, smax_b,
                                                        (float*)d_out);
}